// FLAME_54245436949017
// MI455X (gfx1250) — compile-verified
//
#include <hip/hip_runtime.h>
#include <math.h>

typedef __attribute__((ext_vector_type(2))) float v2f;
typedef __attribute__((ext_vector_type(8))) float v8f;

#define B_N   1024
#define V_N   5023
#define J_N   5
#define NB_N  150
#define NCOLS (V_N * 3)      // 15069
#define KSD   152            // shapedirs K padded to multiple of 4
#define KPF   36             // pose-feature K (exactly 9*4)
#define KTOT  (KSD + KPF)    // 188
#define KSTR  196            // LDS row stride (dwords): conflict-free, 8B-aligned

// workspace layout (float offsets)
#define WS_JT  0                         // 15
#define WS_JS  16                        // 2250  (J*3*NB)
#define WS_PF  4096                      // B*36
#define WS_A   40960                     // B*5*12
// total ~102400 floats = 400 KB

// ---- CDNA5 async global->LDS staging (guarded; falls back to sync fill) ---
#if __has_builtin(__builtin_amdgcn_global_load_async_to_lds_b32)
#define HAVE_ASYNC_LDS 1
typedef __attribute__((address_space(1))) int* as1_i32p;
typedef __attribute__((address_space(3))) int* as3_i32p;
__device__ __forceinline__ void cp_async_f32(float* lds, const float* g) {
    __builtin_amdgcn_global_load_async_to_lds_b32(
        (as1_i32p)g, (as3_i32p)lds, 0, 0);
}
__device__ __forceinline__ void async_wait0() {
#if __has_builtin(__builtin_amdgcn_s_wait_asynccnt)
    __builtin_amdgcn_s_wait_asynccnt(0);
#else
    asm volatile("s_wait_asynccnt 0x0" ::: "memory");
#endif
}
#endif

// ---------------- Kernel 1: fold J_regressor into shapedirs/v_template ----
__global__ void __launch_bounds__(128)
jreg_precompute(const float* __restrict__ Jreg, const float* __restrict__ vt,
                const float* __restrict__ SD, float* __restrict__ ws) {
    __shared__ float red[128];
    int o = blockIdx.x;        // 0..2264 : 15 Jt + 2250 JS
    int tid = threadIdx.x;
    float acc = 0.f;
    if (o < 15) {
        int j = o / 3, k = o % 3;
        for (int v = tid; v < V_N; v += 128)
            acc += Jreg[j * V_N + v] * vt[v * 3 + k];
    } else {
        int o2 = o - 15;
        int j = o2 / (3 * NB_N), rem = o2 % (3 * NB_N);
        int k = rem / NB_N, l = rem % NB_N;
        for (int v = tid; v < V_N; v += 128)
            acc += Jreg[j * V_N + v] * SD[(v * 3 + k) * NB_N + l];
    }
    red[tid] = acc;
    __syncthreads();
    for (int s = 64; s > 0; s >>= 1) {
        if (tid < s) red[tid] += red[tid + s];
        __syncthreads();
    }
    if (tid == 0) {
        if (o < 15) ws[WS_JT + o] = red[0];
        else        ws[WS_JS + (o - 15)] = red[0];
    }
}

// ---------------- Kernel 2: per-batch pose pipeline -----------------------
__global__ void __launch_bounds__(256)
pose_kernel(const float* __restrict__ betas, const float* __restrict__ pose,
            float* ws, float* __restrict__ out_joints) {
    int b = blockIdx.x * blockDim.x + threadIdx.x;
    if (b >= B_N) return;
    const float* Jt = ws + WS_JT;
    const float* JS = ws + WS_JS;

    // Rodrigues (matches reference: epsilon added per-component before norm)
    float R[J_N][9];
    for (int j = 0; j < J_N; ++j) {
        float rx = pose[b * 15 + 3 * j + 0];
        float ry = pose[b * 15 + 3 * j + 1];
        float rz = pose[b * 15 + 3 * j + 2];
        float ex = rx + 1e-8f, ey = ry + 1e-8f, ez = rz + 1e-8f;
        float ang = sqrtf(ex * ex + ey * ey + ez * ez);
        float inv = 1.f / ang;
        float ux = rx * inv, uy = ry * inv, uz = rz * inv;
        float c = cosf(ang), s = sinf(ang), t = 1.f - c;
        float ss = ux * ux + uy * uy + uz * uz;   // K^2 = uu^T - ss*I
        R[j][0] = 1.f + t * (ux * ux - ss);
        R[j][1] = -s * uz + t * (ux * uy);
        R[j][2] =  s * uy + t * (ux * uz);
        R[j][3] =  s * uz + t * (uy * ux);
        R[j][4] = 1.f + t * (uy * uy - ss);
        R[j][5] = -s * ux + t * (uy * uz);
        R[j][6] = -s * uy + t * (uz * ux);
        R[j][7] =  s * ux + t * (uz * uy);
        R[j][8] = 1.f + t * (uz * uz - ss);
    }
    // pose_feature -> ws
    float* pf = ws + WS_PF + b * 36;
    for (int j = 1; j < J_N; ++j)
        for (int i = 0; i < 9; ++i) {
            float id = (i == 0 || i == 4 || i == 8) ? 1.f : 0.f;
            pf[(j - 1) * 9 + i] = R[j][i] - id;
        }
    // joints: Jl = Jt + JS @ betas[b]
    float Jl[J_N][3];
    const float* bb = betas + b * NB_N;
    for (int j = 0; j < J_N; ++j)
        for (int k = 0; k < 3; ++k) {
            float a = Jt[j * 3 + k];
            const float* row = JS + (j * 3 + k) * NB_N;
            for (int l = 0; l < NB_N; ++l) a += row[l] * bb[l];
            Jl[j][k] = a;
        }
    const int par[J_N] = {-1, 0, 1, 1, 1};
    float rel[J_N][3];
    for (int k = 0; k < 3; ++k) rel[0][k] = Jl[0][k];
    for (int j = 1; j < J_N; ++j)
        for (int k = 0; k < 3; ++k) rel[j][k] = Jl[j][k] - Jl[par[j]][k];
    // kinematic chain, 3x4 transforms
    float G[J_N][12];
    for (int r = 0; r < 3; ++r) {
        G[0][r * 4 + 0] = R[0][r * 3 + 0];
        G[0][r * 4 + 1] = R[0][r * 3 + 1];
        G[0][r * 4 + 2] = R[0][r * 3 + 2];
        G[0][r * 4 + 3] = rel[0][r];
    }
    for (int j = 1; j < J_N; ++j) {
        int p = par[j];
        for (int r = 0; r < 3; ++r) {
            for (int cc = 0; cc < 3; ++cc)
                G[j][r * 4 + cc] = G[p][r * 4 + 0] * R[j][0 + cc] +
                                   G[p][r * 4 + 1] * R[j][3 + cc] +
                                   G[p][r * 4 + 2] * R[j][6 + cc];
            G[j][r * 4 + 3] = G[p][r * 4 + 0] * rel[j][0] +
                              G[p][r * 4 + 1] * rel[j][1] +
                              G[p][r * 4 + 2] * rel[j][2] + G[p][r * 4 + 3];
        }
    }
    // outputs: posed joints + relative transforms A (3x4)
    float* Aw = ws + WS_A + b * 60;
    for (int j = 0; j < J_N; ++j) {
        for (int k = 0; k < 3; ++k)
            out_joints[b * 15 + j * 3 + k] = G[j][k * 4 + 3];
        for (int r = 0; r < 3; ++r) {
            float rj = G[j][r * 4 + 0] * Jl[j][0] + G[j][r * 4 + 1] * Jl[j][1] +
                       G[j][r * 4 + 2] * Jl[j][2];
            Aw[j * 12 + r * 4 + 0] = G[j][r * 4 + 0];
            Aw[j * 12 + r * 4 + 1] = G[j][r * 4 + 1];
            Aw[j * 12 + r * 4 + 2] = G[j][r * 4 + 2];
            Aw[j * 12 + r * 4 + 3] = G[j][r * 4 + 3] - rj;
        }
    }
}

// ---------------- Kernel 3: fused WMMA GEMM -------------------------------
// v_posed[b,n] = vt[n] + sum_l betas[b,l]*SD[n,l] + sum_p pf[b,p]*PD[p,n]
__global__ void __launch_bounds__(128)
gemm_kernel(const float* __restrict__ betas, const float* __restrict__ SD,
            const float* __restrict__ PD, const float* __restrict__ vt,
            const float* __restrict__ ws, float* __restrict__ verts) {
    __shared__ float lA[16 * KSTR];       // betas||pf tile (16 batches)
    __shared__ float lB[4][16 * KSTR];    // SD||PD^T tiles (4 x 16 n-rows)
    int tid = threadIdx.x;
    int mbase = blockIdx.y * 16;
    int nbase0 = blockIdx.x * 64;
    const float* pf = ws + WS_PF;

#ifdef HAVE_ASYNC_LDS
    for (int idx = tid; idx < 16 * KTOT; idx += 128) {
        int m = idx / KTOT, kk = idx % KTOT;
        int gb = mbase + m;
        float* dst = &lA[m * KSTR + kk];
        if (kk < NB_N)      cp_async_f32(dst, &betas[gb * NB_N + kk]);
        else if (kk < KSD)  *dst = 0.f;
        else                cp_async_f32(dst, &pf[gb * 36 + (kk - KSD)]);
    }
    for (int idx = tid; idx < 64 * KTOT; idx += 128) {
        int r = idx / KTOT, kk = idx % KTOT;
        int n = nbase0 + r;
        float* dst = &lB[r >> 4][(r & 15) * KSTR + kk];
        if (n >= NCOLS || (kk >= NB_N && kk < KSD)) *dst = 0.f;
        else if (kk < NB_N) cp_async_f32(dst, &SD[n * NB_N + kk]);
        else                cp_async_f32(dst, &PD[(kk - KSD) * NCOLS + n]);
    }
    async_wait0();
#else
    for (int idx = tid; idx < 16 * KTOT; idx += 128) {
        int m = idx / KTOT, kk = idx % KTOT;
        int gb = mbase + m;
        float val;
        if (kk < NB_N)      val = betas[gb * NB_N + kk];
        else if (kk < KSD)  val = 0.f;
        else                val = pf[gb * 36 + (kk - KSD)];
        lA[m * KSTR + kk] = val;
    }
    for (int idx = tid; idx < 64 * KTOT; idx += 128) {
        int r = idx / KTOT, kk = idx % KTOT;
        int n = nbase0 + r;
        float val = 0.f;
        if (n < NCOLS) {
            if (kk < NB_N)      val = SD[n * NB_N + kk];
            else if (kk < KSD)  val = 0.f;
            else                val = PD[(kk - KSD) * NCOLS + n];
        }
        lB[r >> 4][(r & 15) * KSTR + kk] = val;
    }
#endif
    __syncthreads();

    int wave = tid >> 5;
    int lane = tid & 31;
    int l16 = lane & 15;
    int half = lane >> 4;
    const float* Abase = lA + l16 * KSTR + 2 * half;
    const float* Bbase = &lB[wave][l16 * KSTR + 2 * half];

    v8f c = {0.f, 0.f, 0.f, 0.f, 0.f, 0.f, 0.f, 0.f};
#pragma unroll
    for (int kk = 0; kk < KTOT; kk += 4) {
        v2f a = *(const v2f*)(Abase + kk);   // A[m=l16][k = kk+2*half, +1]
        v2f bf = *(const v2f*)(Bbase + kk);  // B[k][n=l16] (B = SD2^T)
        c = __builtin_amdgcn_wmma_f32_16x16x4_f32(
                false, a, false, bf, (short)0, c, false, false);
    }

    int ncol = nbase0 + wave * 16 + l16;
    if (ncol < NCOLS) {
        float t = vt[ncol];
#pragma unroll
        for (int g = 0; g < 8; ++g) {
            int row = mbase + g + 8 * half;   // D: VGPR g -> M = g + 8*half
            verts[row * NCOLS + ncol] = c[g] + t;
        }
    }
}

// ---------------- Kernel 4: in-place LBS ----------------------------------
__global__ void __launch_bounds__(256)
lbs_kernel(const float* __restrict__ lbsw, const float* __restrict__ ws,
           float* __restrict__ verts) {
    __shared__ float Al[60];
    int bi = blockIdx.x;
    int b = bi / 20, chunk = bi % 20;
    int tid = threadIdx.x;
    if (tid < 60) Al[tid] = ws[WS_A + b * 60 + tid];
    __syncthreads();
    int v = chunk * 256 + tid;
    if (v >= V_N) return;
    size_t base = (size_t)b * NCOLS + v * 3;
    float x = verts[base + 0], y = verts[base + 1], z = verts[base + 2];
    float ox = 0.f, oy = 0.f, oz = 0.f;
#pragma unroll
    for (int j = 0; j < J_N; ++j) {
        float w = lbsw[v * J_N + j];
        const float* A = Al + j * 12;
        ox += w * (A[0] * x + A[1] * y + A[2]  * z + A[3]);
        oy += w * (A[4] * x + A[5] * y + A[6]  * z + A[7]);
        oz += w * (A[8] * x + A[9] * y + A[10] * z + A[11]);
    }
    verts[base + 0] = ox;
    verts[base + 1] = oy;
    verts[base + 2] = oz;
}

// ---------------- host launcher -------------------------------------------
extern "C" void kernel_launch(void* const* d_in, const int* in_sizes, int n_in,
                              void* d_out, int out_size, void* d_ws,
                              size_t ws_size, hipStream_t stream) {
    (void)in_sizes; (void)n_in; (void)out_size; (void)ws_size;
    const float* betas      = (const float*)d_in[0];
    const float* pose       = (const float*)d_in[1];
    const float* v_template = (const float*)d_in[2];
    const float* shapedirs  = (const float*)d_in[3];
    const float* posedirs   = (const float*)d_in[4];
    const float* Jreg       = (const float*)d_in[5];
    const float* lbsw       = (const float*)d_in[6];
    float* out = (float*)d_out;
    float* ws  = (float*)d_ws;
    float* verts  = out;
    float* joints = out + (size_t)B_N * NCOLS;

    jreg_precompute<<<2265, 128, 0, stream>>>(Jreg, v_template, shapedirs, ws);
    pose_kernel<<<(B_N + 255) / 256, 256, 0, stream>>>(betas, pose, ws, joints);
    dim3 gc((NCOLS + 63) / 64, B_N / 16);   // 236 x 64
    gemm_kernel<<<gc, 128, 0, stream>>>(betas, shapedirs, posedirs, v_template,
                                        ws, verts);
    lbs_kernel<<<B_N * 20, 256, 0, stream>>>(lbsw, ws, verts);
}